// TrajectronPooling_63410897158187
// MI455X (gfx1250) — compile-verified
//
#include <hip/hip_runtime.h>
#include <hip/hip_bf16.h>

#ifndef __has_builtin
#define __has_builtin(x) 0
#endif

// ---------------------------------------------------------------------------
// Trajectron pooling, fused for MI455X (gfx1250, wave32, WMMA).
// FP32 end-to-end. HBM floor ~220 MB (~9.4 us @ 23.3 TB/s). 48 rows/block
// amortizes the 1.18 MB packed-weight panel over 3 row-tiles per B-fragment
// load, cutting L2 weight traffic 3x vs 16-row blocks. h staged to LDS via
// gfx1250 async global->LDS copies (ASYNCcnt) when available.
// ---------------------------------------------------------------------------

typedef float v2f __attribute__((ext_vector_type(2)));
typedef float v8f __attribute__((ext_vector_type(8)));
typedef int   v4i __attribute__((ext_vector_type(4)));

#define HID   256
#define NOUT  32
#define KCAT  288          // 32 (emb) + 256 (h)
#define ROWS  48           // rows per block = 3 row-tiles of 16
#define A_STRIDE 292       // 288 + 4 pad -> conflict-free f32 fragment reads

#if __has_builtin(__builtin_amdgcn_global_load_async_to_lds_b128)
#define HAVE_ASYNC_LDS 1
// Builtin signature (from clang diagnostic): (v4i AS1*, v4i AS3*, imm, imm)
typedef __attribute__((address_space(1))) v4i* g1v4i_t;
typedef __attribute__((address_space(3))) v4i* l3v4i_t;
#else
#define HAVE_ASYNC_LDS 0
#endif

static __device__ __forceinline__ void wait_async0() {
#if __has_builtin(__builtin_amdgcn_s_wait_asynccnt)
  __builtin_amdgcn_s_wait_asynccnt(0);
#else
  asm volatile("s_wait_asynccnt 0" ::: "memory");
#endif
}

// D = A(16x4 f32) * B(4x16 f32) + C(16x16 f32)
static __device__ __forceinline__ v8f wmma4(v2f a, v2f b, v8f c) {
  return __builtin_amdgcn_wmma_f32_16x16x4_f32(false, a, false, b, (short)0, c,
                                               false, false);
}

// Branch-free activations: v_exp_f32 + v_rcp_f32, correct +-inf saturation.
static __device__ __forceinline__ float fast_sigmoid(float x) {
  return __builtin_amdgcn_rcpf(1.0f + __expf(-x));
}
static __device__ __forceinline__ float fast_tanh(float x) {
  return 1.0f - 2.0f * __builtin_amdgcn_rcpf(__expf(2.0f * x) + 1.0f);
}

// ---------------------------------------------------------------------------
// Kernel 0: zero the reduction slots in workspace.
// ---------------------------------------------------------------------------
__global__ void tp_init_ws(float* wsf) {
  if (threadIdx.x < 4) wsf[threadIdx.x] = 0.0f;
  if (threadIdx.x == 4) ((int*)wsf)[4] = 0;
}

// ---------------------------------------------------------------------------
// Kernel 1: totals = sum over rows of [pos, vel]; count = sum(track_mask).
// ---------------------------------------------------------------------------
__global__ __launch_bounds__(256) void tp_reduce(
    const float* __restrict__ obs1, const float* __restrict__ obs2,
    const unsigned char* __restrict__ mask, float* __restrict__ totals,
    int* __restrict__ cnt, int N) {
  __shared__ float s0[256], s1[256], s2[256], s3[256];
  __shared__ int si[256];
  int t = threadIdx.x;
  int i = blockIdx.x * 256 + t;
  float a = 0.f, b = 0.f, c = 0.f, d = 0.f;
  int m = 0;
  if (i < N) {
    float o1x = obs1[2 * i], o1y = obs1[2 * i + 1];
    float o2x = obs2[2 * i], o2y = obs2[2 * i + 1];
    a = o2x; b = o2y; c = o2x - o1x; d = o2y - o1y;
    m = mask[i] ? 1 : 0;
  }
  s0[t] = a; s1[t] = b; s2[t] = c; s3[t] = d; si[t] = m;
  __syncthreads();
  for (int s = 128; s > 0; s >>= 1) {
    if (t < s) {
      s0[t] += s0[t + s]; s1[t] += s1[t + s];
      s2[t] += s2[t + s]; s3[t] += s3[t + s];
      si[t] += si[t + s];
    }
    __syncthreads();
  }
  if (t == 0) {
    atomicAdd(&totals[0], s0[0]);
    atomicAdd(&totals[1], s1[0]);
    atomicAdd(&totals[2], s2[0]);
    atomicAdd(&totals[3], s3[0]);
    atomicAdd(cnt, si[0]);
  }
}

// ---------------------------------------------------------------------------
// Kernel 2: pack W_cat[1024][288] = [W_ih | W_hh]; b_cat = b_ih + b_hh.
// ---------------------------------------------------------------------------
__global__ __launch_bounds__(256) void tp_pack(
    const float* __restrict__ W_ih, const float* __restrict__ b_ih,
    const float* __restrict__ W_hh, const float* __restrict__ b_hh,
    float* __restrict__ W_cat, float* __restrict__ b_cat) {
  int idx = blockIdx.x * 256 + threadIdx.x;
  if (idx < 4 * HID * KCAT) {
    int j = idx / KCAT, k = idx % KCAT;
    W_cat[idx] = (k < NOUT) ? W_ih[j * NOUT + k] : W_hh[j * HID + (k - NOUT)];
  }
  if (idx < 4 * HID) b_cat[idx] = b_ih[idx] + b_hh[idx];
}

// ---------------------------------------------------------------------------
// Kernel 3: fused emb -> gate GEMM (WMMA) -> LSTM elementwise -> out GEMM.
// 48 rows per workgroup, 8 waves. Single LDS panel: [emb(32) | h(256)] per
// row; h_new is written back into the h region after a barrier.
// ---------------------------------------------------------------------------
__global__ __launch_bounds__(256) void tp_fused(
    const float* __restrict__ obs1, const float* __restrict__ obs2,
    const float* __restrict__ h, const float* __restrict__ c,
    const float* __restrict__ W_emb, const float* __restrict__ b_emb,
    const float* __restrict__ W_out, const float* __restrict__ b_out,
    const float* __restrict__ W_cat, const float* __restrict__ b_cat,
    const float* __restrict__ totals, const int* __restrict__ cnt,
    float* __restrict__ out, int N) {
  __shared__ float Abuf[ROWS * A_STRIDE];  // 56,064 B

  const int tid  = threadIdx.x;
  const int wave = tid >> 5;
  const int lane = tid & 31;
  const int hf   = lane >> 4;   // 0: K=k0..k0+1, 1: K=k0+2..k0+3
  const int l16  = lane & 15;
  const int row0 = blockIdx.x * ROWS;

  // ---- Phase 1a: stage h rows into LDS (async global->LDS if available) --
  for (int idx = tid; idx < ROWS * (HID / 4); idx += 256) {
    int i = idx >> 6, k4 = idx & 63;
    int row = row0 + i; row = row < N ? row : N - 1;
    const float* g = h + (size_t)row * HID + 4 * k4;
    float* l = &Abuf[i * A_STRIDE + NOUT + 4 * k4];
#if HAVE_ASYNC_LDS
    __builtin_amdgcn_global_load_async_to_lds_b128((g1v4i_t)(void*)g,
                                                   (l3v4i_t)(void*)l, 0, 0);
#else
    float4 hv = *reinterpret_cast<const float4*>(g);
    l[0] = hv.x; l[1] = hv.y; l[2] = hv.z; l[3] = hv.w;
#endif
  }

  // ---- Phase 1b: embedding columns ---------------------------------------
  const float T0 = totals[0], T1 = totals[1], T2 = totals[2], T3 = totals[3];
  for (int idx = tid; idx < ROWS * NOUT; idx += 256) {
    int i = idx >> 5, j = idx & 31;
    int row = row0 + i; row = row < N ? row : N - 1;
    float o1x = obs1[2 * row], o1y = obs1[2 * row + 1];
    float o2x = obs2[2 * row], o2y = obs2[2 * row + 1];
    float vx = o2x - o1x, vy = o2y - o1y;
    const float* wr = W_emb + j * 8;
    float s = b_emb[j];
    s += o2x * wr[0] + o2y * wr[1] + vx * wr[2] + vy * wr[3];
    s += (T0 - o2x) * wr[4] + (T1 - o2y) * wr[5] +
         (T2 - vx) * wr[6] + (T3 - vy) * wr[7];
    Abuf[i * A_STRIDE + j] = fmaxf(s, 0.0f);
  }
#if HAVE_ASYNC_LDS
  wait_async0();
#endif
  __syncthreads();

  // ---- Phase 2: gates = A(48x288) x W_cat^T. Each wave: 2 hidden tiles,
  //      4 gate tiles x 3 row-tiles per B-fragment (3x B reuse). -----------
  const float* Ar0 = &Abuf[(0 * 16 + l16) * A_STRIDE];
  const float* Ar1 = &Abuf[(1 * 16 + l16) * A_STRIDE];
  const float* Ar2 = &Abuf[(2 * 16 + l16) * A_STRIDE];
  float hn[2][3][8];  // h_new kept in registers across the reuse barrier

  for (int t = 0; t < 2; ++t) {
    const int u0 = (wave * 2 + t) * 16;  // hidden-unit tile base
    v8f acc[3][4];
#pragma unroll
    for (int rt = 0; rt < 3; ++rt)
#pragma unroll
      for (int g = 0; g < 4; ++g) acc[rt][g] = (v8f){};

    const float* B0 = W_cat + (size_t)(0 * HID + u0 + l16) * KCAT;
    const float* B1 = W_cat + (size_t)(1 * HID + u0 + l16) * KCAT;
    const float* B2 = W_cat + (size_t)(2 * HID + u0 + l16) * KCAT;
    const float* B3 = W_cat + (size_t)(3 * HID + u0 + l16) * KCAT;

#pragma unroll 2
    for (int k0 = 0; k0 < KCAT; k0 += 4) {
      const int kk = k0 + 2 * hf;
      v2f a0 = *(const v2f*)(Ar0 + kk);
      v2f a1 = *(const v2f*)(Ar1 + kk);
      v2f a2 = *(const v2f*)(Ar2 + kk);
      v2f b0 = *(const v2f*)(B0 + kk);
      v2f b1 = *(const v2f*)(B1 + kk);
      v2f b2 = *(const v2f*)(B2 + kk);
      v2f b3 = *(const v2f*)(B3 + kk);
      acc[0][0] = wmma4(a0, b0, acc[0][0]);
      acc[1][0] = wmma4(a1, b0, acc[1][0]);
      acc[2][0] = wmma4(a2, b0, acc[2][0]);
      acc[0][1] = wmma4(a0, b1, acc[0][1]);
      acc[1][1] = wmma4(a1, b1, acc[1][1]);
      acc[2][1] = wmma4(a2, b1, acc[2][1]);
      acc[0][2] = wmma4(a0, b2, acc[0][2]);
      acc[1][2] = wmma4(a1, b2, acc[1][2]);
      acc[2][2] = wmma4(a2, b2, acc[2][2]);
      acc[0][3] = wmma4(a0, b3, acc[0][3]);
      acc[1][3] = wmma4(a1, b3, acc[1][3]);
      acc[2][3] = wmma4(a2, b3, acc[2][3]);
    }

    const float bi = b_cat[0 * HID + u0 + l16];
    const float bf = b_cat[1 * HID + u0 + l16];
    const float bg = b_cat[2 * HID + u0 + l16];
    const float bo = b_cat[3 * HID + u0 + l16];
#pragma unroll
    for (int rt = 0; rt < 3; ++rt) {
#pragma unroll
      for (int r = 0; r < 8; ++r) {
        const int M = r + hf * 8;  // C/D layout: M = r + 8*half
        int row = row0 + rt * 16 + M; row = row < N ? row : N - 1;
        const float cold = c[(size_t)row * HID + u0 + l16];
        const float ig = fast_sigmoid(acc[rt][0][r] + bi);
        const float fg = fast_sigmoid(acc[rt][1][r] + bf);
        const float gg = fast_tanh(acc[rt][2][r] + bg);
        const float og = fast_sigmoid(acc[rt][3][r] + bo);
        const float cn = fg * cold + ig * gg;
        hn[t][rt][r] = og * fast_tanh(cn);
      }
    }
  }

  __syncthreads();  // everyone done reading old h from Abuf
#pragma unroll
  for (int t = 0; t < 2; ++t) {
    const int u0 = (wave * 2 + t) * 16;
#pragma unroll
    for (int rt = 0; rt < 3; ++rt)
#pragma unroll
      for (int r = 0; r < 8; ++r) {
        const int M = r + hf * 8;
        Abuf[(rt * 16 + M) * A_STRIDE + NOUT + u0 + l16] = hn[t][rt][r];
      }
  }
  __syncthreads();  // h_new panel visible to all waves

  // ---- Phase 3: out = h_new(48x256) x W_out^T (6 tiles, waves 0..5) ------
  if (wave < 6) {
    const int rt = wave >> 1;
    const int c0 = (wave & 1) * 16;
    v8f acc = (v8f){};
    const float* Hr = &Abuf[(rt * 16 + l16) * A_STRIDE + NOUT];
    const float* Br = W_out + (size_t)(c0 + l16) * HID;
#pragma unroll 4
    for (int k0 = 0; k0 < HID; k0 += 4) {
      const int kk = k0 + 2 * hf;
      v2f a = *(const v2f*)(Hr + kk);
      v2f b = *(const v2f*)(Br + kk);
      acc = wmma4(a, b, acc);
    }
    const float bb = b_out[c0 + l16];
    const bool zero = (*cnt == 1);  // only-one-track short-circuit
#pragma unroll
    for (int r = 0; r < 8; ++r) {
      const int M = r + hf * 8;
      const int row = row0 + rt * 16 + M;
      if (row < N) out[(size_t)row * NOUT + c0 + l16] = zero ? 0.0f : acc[r] + bb;
    }
  }
}

// ---------------------------------------------------------------------------
extern "C" void kernel_launch(void* const* d_in, const int* in_sizes, int n_in,
                              void* d_out, int out_size, void* d_ws, size_t ws_size,
                              hipStream_t stream) {
  const float* obs1  = (const float*)d_in[0];
  const float* obs2  = (const float*)d_in[1];
  const float* h     = (const float*)d_in[2];
  const float* c     = (const float*)d_in[3];
  const unsigned char* mask = (const unsigned char*)d_in[4];  // bool mask
  const float* W_emb = (const float*)d_in[5];
  const float* b_emb = (const float*)d_in[6];
  const float* W_ih  = (const float*)d_in[7];
  const float* b_ih  = (const float*)d_in[8];
  const float* W_hh  = (const float*)d_in[9];
  const float* b_hh  = (const float*)d_in[10];
  const float* W_out = (const float*)d_in[11];
  const float* b_out = (const float*)d_in[12];
  float* out = (float*)d_out;
  const int N = in_sizes[0] / 2;

  // Workspace (floats): [0..3] totals, [4] mask count (int),
  // [8..1031] b_cat, [1032..] W_cat (1024x288).  ~1.19 MB total.
  float* wsf    = (float*)d_ws;
  float* totals = wsf;
  int*   cnt    = (int*)(wsf + 4);
  float* b_cat  = wsf + 8;
  float* W_cat  = wsf + 8 + 4 * HID;

  tp_init_ws<<<1, 32, 0, stream>>>(wsf);
  tp_reduce<<<(N + 255) / 256, 256, 0, stream>>>(obs1, obs2, mask, totals, cnt, N);
  tp_pack<<<(4 * HID * KCAT + 255) / 256, 256, 0, stream>>>(W_ih, b_ih, W_hh, b_hh,
                                                            W_cat, b_cat);
  tp_fused<<<(N + ROWS - 1) / ROWS, 256, 0, stream>>>(obs1, obs2, h, c,
                                                      W_emb, b_emb, W_out, b_out,
                                                      W_cat, b_cat, totals, cnt,
                                                      out, N);
}